// SinkhornDistance_1838246003013
// MI455X (gfx1250) — compile-verified
//
#include <hip/hip_runtime.h>
#include <hip/hip_bf16.h>

// ---------------- problem constants (match reference shapes) ----------------
#define BATCH   8
#define NN      2048          // rows per batch
#define DHW     49            // 7*7
#define LDP     52            // padded row length for softmax outputs (even, >=49)
#define EPSV    1e-3f
#define NITER   100
#define NSPLIT  8             // n-chunks for transposed GEMV partials

typedef __attribute__((ext_vector_type(2))) float v2f;
typedef __attribute__((ext_vector_type(8))) float v8f;

// ---------------------------------------------------------------------------
// Kernel 1: row softmax over 49 elems + squared-norm of the softmax row.
// One wave32 per row; lane L covers elements L and L+32. Pads cols 49..51 = 0.
// ---------------------------------------------------------------------------
__global__ void sk_softmax_rows(const float* __restrict__ in,
                                float* __restrict__ outp,   // [rows][LDP]
                                float* __restrict__ sq,     // [rows]
                                int nrows)
{
    int wid  = (blockIdx.x * blockDim.x + threadIdx.x) >> 5;
    int lane = threadIdx.x & 31;
    if (wid >= nrows) return;

    const float* r = in + (size_t)wid * DHW;
    float x0 = r[lane];                                    // lane 0..31 < 49
    float x1 = (lane + 32 < DHW) ? r[lane + 32] : -3.0e38f;

    float m = fmaxf(x0, x1);
    #pragma unroll
    for (int off = 16; off > 0; off >>= 1) m = fmaxf(m, __shfl_xor(m, off, 32));

    float e0 = __expf(x0 - m);
    float e1 = (lane + 32 < DHW) ? __expf(x1 - m) : 0.0f;

    float s = e0 + e1;
    #pragma unroll
    for (int off = 16; off > 0; off >>= 1) s += __shfl_xor(s, off, 32);

    float inv = 1.0f / s;
    float p0 = e0 * inv;
    float p1 = e1 * inv;

    float q = p0 * p0 + p1 * p1;
    #pragma unroll
    for (int off = 16; off > 0; off >>= 1) q += __shfl_xor(q, off, 32);

    float* o = outp + (size_t)wid * LDP;
    o[lane] = p0;                       // cols 0..31
    if (lane + 32 < LDP) o[lane + 32] = p1;   // cols 32..51 (49..51 are zero)
    if (lane == 0) sq[wid] = q;
}

// ---------------------------------------------------------------------------
// Kernel 2: C[b,n,m] = sqrt(max(a2[n] + b2[m] - 2*(xf[n,:]·yf[m,:]), 1e-12))
// One wave32 per 16x16 output tile; FP32 WMMA 16x16x4, K padded 49->52.
// A-lane layout (32-bit A 16x4): lane<16 -> K=0,1 ; lane>=16 -> K=2,3.
// B = yf^T has the identical per-lane load pattern from row-major yf.
// ---------------------------------------------------------------------------
__global__ void sk_cdist_wmma(const float* __restrict__ xf,
                              const float* __restrict__ yf,
                              const float* __restrict__ a2,
                              const float* __restrict__ b2,
                              float* __restrict__ C)
{
    const int tilesN   = NN / 16;               // 128
    const int tilesPer = tilesN * tilesN;       // per batch

    int wid  = blockIdx.x * (blockDim.x >> 5) + (threadIdx.x >> 5);
    int lane = threadIdx.x & 31;

    int b  = wid / tilesPer;
    int t  = wid - b * tilesPer;
    int tn = t / tilesN;
    int tm = t - tn * tilesN;
    if (b >= BATCH) return;                     // wave-uniform; EXEC stays full

    const float* Ap = xf + ((size_t)b * NN + tn * 16 + (lane & 15)) * LDP + ((lane >> 4) * 2);
    const float* Bp = yf + ((size_t)b * NN + tm * 16 + (lane & 15)) * LDP + ((lane >> 4) * 2);

    v8f acc = {};
    #pragma unroll
    for (int d0 = 0; d0 < LDP; d0 += 4) {
        v2f a = *(const v2f*)(Ap + d0);
        v2f bb = *(const v2f*)(Bp + d0);
        acc = __builtin_amdgcn_wmma_f32_16x16x4_f32(
                  false, a, false, bb, (short)0, acc, false, false);
    }

    // D layout: VGPR r, lane<16 -> row r, col lane ; lane>=16 -> row r+8, col lane-16
    int col   = lane & 15;
    int rbase = (lane >> 4) * 8;
    float bn2 = b2[(size_t)b * NN + tm * 16 + col];
    float* Cb = C + ((size_t)b * NN + tn * 16) * NN + (size_t)tm * 16 + col;

    #pragma unroll
    for (int r = 0; r < 8; ++r) {
        int row  = rbase + r;
        float an2 = a2[(size_t)b * NN + tn * 16 + row];
        float val = an2 + bn2 - 2.0f * acc[r];
        Cb[(size_t)row * NN] = __builtin_sqrtf(fmaxf(val, 1e-12f));
    }
}

// ---------------------------------------------------------------------------
// Kernel 3: u[b,n] = EPS / (sum_m C[b,n,m]*v[b,m] + EPS)
// One wave32 per row, float4 per lane -> 128B coalesced, L2-resident streaming.
// ---------------------------------------------------------------------------
__global__ void sk_row_gemv_u(const float* __restrict__ C,
                              const float* __restrict__ v,
                              float* __restrict__ u)
{
    int wid  = blockIdx.x * (blockDim.x >> 5) + (threadIdx.x >> 5);
    int lane = threadIdx.x & 31;
    int b = wid / NN;
    int n = wid - b * NN;

    const float4* r4 = (const float4*)(C + ((size_t)b * NN + n) * NN);
    const float4* v4 = (const float4*)(v + (size_t)b * NN);

    float s = 0.0f;
    #pragma unroll 4
    for (int m = lane; m < NN / 4; m += 32) {
        float4 c = r4[m];
        float4 w = v4[m];
        s = fmaf(c.x, w.x, fmaf(c.y, w.y, fmaf(c.z, w.z, fmaf(c.w, w.w, s))));
    }
    #pragma unroll
    for (int off = 16; off > 0; off >>= 1) s += __shfl_xor(s, off, 32);

    if (lane == 0) u[(size_t)b * NN + n] = EPSV / (s + EPSV);
}

// ---------------------------------------------------------------------------
// Kernel 4a: partial sums for v-update: part[k][b,m] = sum_{n in chunk k} C[b,n,m]*u[b,n]
// Lane-per-column -> consecutive lanes read consecutive addresses each n step.
// Deterministic (no atomics).
// ---------------------------------------------------------------------------
__global__ void sk_col_gemv_part(const float* __restrict__ C,
                                 const float* __restrict__ u,
                                 float* __restrict__ part)
{
    int id = blockIdx.x * blockDim.x + threadIdx.x;   // flat over BATCH*NN
    int b = id / NN;
    int m = id - b * NN;

    const float* base = C + (size_t)b * NN * NN + m;
    const float* ub   = u + (size_t)b * NN;

    const int chunk = NN / NSPLIT;
    int n0 = blockIdx.y * chunk;

    float s = 0.0f;
    #pragma unroll 4
    for (int n = n0; n < n0 + chunk; ++n)
        s = fmaf(base[(size_t)n * NN], ub[n], s);

    part[(size_t)blockIdx.y * (BATCH * NN) + id] = s;
}

// Kernel 4b: reduce partials deterministically, apply EPS/(x+EPS)
__global__ void sk_col_reduce_v(const float* __restrict__ part,
                                float* __restrict__ v)
{
    int id = blockIdx.x * blockDim.x + threadIdx.x;
    float s = 0.0f;
    #pragma unroll
    for (int k = 0; k < NSPLIT; ++k) s += part[(size_t)k * (BATCH * NN) + id];
    v[id] = EPSV / (s + EPSV);
}

// ---------------------------------------------------------------------------
// Kernel 5: init u,v to 1/N
// ---------------------------------------------------------------------------
__global__ void sk_init_uv(float* __restrict__ u, float* __restrict__ v)
{
    int id = blockIdx.x * blockDim.x + threadIdx.x;
    u[id] = 1.0f / (float)NN;
    v[id] = 1.0f / (float)NN;
}

// ---------------------------------------------------------------------------
// Kernel 6a: rowvals[b,n] = u[b,n] * (sum_m C[b,n,m]*v[b,m])
// ---------------------------------------------------------------------------
__global__ void sk_final_rows(const float* __restrict__ C,
                              const float* __restrict__ u,
                              const float* __restrict__ v,
                              float* __restrict__ rowvals)
{
    int wid  = blockIdx.x * (blockDim.x >> 5) + (threadIdx.x >> 5);
    int lane = threadIdx.x & 31;
    int b = wid / NN;
    int n = wid - b * NN;

    const float4* r4 = (const float4*)(C + ((size_t)b * NN + n) * NN);
    const float4* v4 = (const float4*)(v + (size_t)b * NN);

    float s = 0.0f;
    #pragma unroll 4
    for (int m = lane; m < NN / 4; m += 32) {
        float4 c = r4[m];
        float4 w = v4[m];
        s = fmaf(c.x, w.x, fmaf(c.y, w.y, fmaf(c.z, w.z, fmaf(c.w, w.w, s))));
    }
    #pragma unroll
    for (int off = 16; off > 0; off >>= 1) s += __shfl_xor(s, off, 32);

    if (lane == 0) rowvals[(size_t)b * NN + n] = u[(size_t)b * NN + n] * s;
}

// Kernel 6b: single-block deterministic reduction, mean over batch
__global__ void sk_final_reduce(const float* __restrict__ rowvals,
                                float* __restrict__ out)
{
    __shared__ float sm[256];
    const int total = BATCH * NN;
    float s = 0.0f;
    for (int i = threadIdx.x; i < total; i += 256) s += rowvals[i];
    sm[threadIdx.x] = s;
    __syncthreads();
    #pragma unroll
    for (int k = 128; k > 0; k >>= 1) {
        if (threadIdx.x < (unsigned)k) sm[threadIdx.x] += sm[threadIdx.x + k];
        __syncthreads();
    }
    if (threadIdx.x == 0) out[0] = sm[0] * (1.0f / (float)BATCH);
}

// ---------------------------------------------------------------------------
extern "C" void kernel_launch(void* const* d_in, const int* in_sizes, int n_in,
                              void* d_out, int out_size, void* d_ws, size_t ws_size,
                              hipStream_t stream)
{
    (void)in_sizes; (void)n_in; (void)out_size; (void)ws_size;

    const float* x = (const float*)d_in[0];
    const float* y = (const float*)d_in[1];
    float* out = (float*)d_out;

    // ---- workspace layout (256B aligned slices) ----
    const size_t ROWS = (size_t)BATCH * NN;        // 16384
    auto align256 = [](size_t v) { return (v + 255) & ~(size_t)255; };

    char* ws = (char*)d_ws;
    size_t off = 0;
    float* xf   = (float*)(ws + off); off += align256(ROWS * LDP * sizeof(float));
    float* yf   = (float*)(ws + off); off += align256(ROWS * LDP * sizeof(float));
    float* a2   = (float*)(ws + off); off += align256(ROWS * sizeof(float));
    float* b2   = (float*)(ws + off); off += align256(ROWS * sizeof(float));
    float* u    = (float*)(ws + off); off += align256(ROWS * sizeof(float));
    float* v    = (float*)(ws + off); off += align256(ROWS * sizeof(float));
    float* part = (float*)(ws + off); off += align256((size_t)NSPLIT * ROWS * sizeof(float));
    float* rowvals = (float*)(ws + off); off += align256(ROWS * sizeof(float));
    float* C    = (float*)(ws + off); off += align256(ROWS * (size_t)NN * sizeof(float)); // 134 MB

    // ---- 1) softmax rows (+ squared norms) ----
    {
        dim3 blk(256);
        dim3 grd((unsigned)(ROWS / 8));            // 8 waves per block, 1 wave per row
        sk_softmax_rows<<<grd, blk, 0, stream>>>(x, xf, a2, (int)ROWS);
        sk_softmax_rows<<<grd, blk, 0, stream>>>(y, yf, b2, (int)ROWS);
    }

    // ---- 2) C = cdist via FP32 WMMA ----
    {
        const int tilesN = NN / 16;
        const size_t waves = (size_t)BATCH * tilesN * tilesN;   // 131072
        dim3 blk(256);                                          // 8 waves
        dim3 grd((unsigned)(waves / 8));
        sk_cdist_wmma<<<grd, blk, 0, stream>>>(xf, yf, a2, b2, C);
    }

    // ---- 3) init u,v ----
    sk_init_uv<<<dim3((unsigned)(ROWS / 256)), dim3(256), 0, stream>>>(u, v);

    // ---- 4) 100 Sinkhorn iterations ----
    {
        dim3 blkR(256), grdR((unsigned)(ROWS / 8));             // row GEMV: wave/row
        dim3 blkC(256), grdC((unsigned)(ROWS / 256), NSPLIT);   // col partials
        dim3 blkV(256), grdV((unsigned)(ROWS / 256));           // reduce
        for (int it = 0; it < NITER; ++it) {
            sk_row_gemv_u<<<grdR, blkR, 0, stream>>>(C, v, u);
            sk_col_gemv_part<<<grdC, blkC, 0, stream>>>(C, u, part);
            sk_col_reduce_v<<<grdV, blkV, 0, stream>>>(part, v);
        }
    }

    // ---- 5) distance = mean_b( u^T C v ) ----
    sk_final_rows<<<dim3((unsigned)(ROWS / 8)), dim3(256), 0, stream>>>(C, u, v, rowvals);
    sk_final_reduce<<<dim3(1), dim3(256), 0, stream>>>(rowvals, out);
}